// MattingLaplacianLoss_43370579755260
// MI455X (gfx1250) — compile-verified
//
#include <hip/hip_runtime.h>

// Problem constants (reference: B=4, C=3, H=W=256, 3x3 windows)
#define BATCH   4
#define HH      256
#define WW      256
#define HP      (HH - 2)               // 254
#define WP      (WW - 2)               // 254
#define NPATCH  (HP * WP)              // 64516
#define TOTALP  (BATCH * NPATCH)       // 258064
#define NBANDS  25
#define CENTER  12
#define EPSF    1e-7f
#define NPIX    (HH * WW)              // 65536

#define TPB         128                // 4 waves per block, thread-per-patch setup
#define SLOT_STRIDE 109                // 108 payload floats + 1 pad; odd => bank-conflict-free

typedef float v2f __attribute__((ext_vector_type(2)));
typedef float v8f __attribute__((ext_vector_type(8)));

// ---------------------------------------------------------------------------
// Pass 0: zero the band workspace and the scalar output.
// ---------------------------------------------------------------------------
__global__ void zero_kernel(float* __restrict__ wb, long n, float* __restrict__ out) {
    long i = (long)blockIdx.x * blockDim.x + threadIdx.x;
    if (i < n) wb[i] = 0.0f;
    if (i == 0) out[0] = 0.0f;
}

// ---------------------------------------------------------------------------
// Thread-per-patch setup: load 3x3x3 window, mean / covariance / symmetric
// 3x3 inverse, then stage t = (sgn/9) * (diff * IC)  (27 floats, WMMA A rows)
// and diff (27 floats, WMMA B cols) into LDS.  sgn=-1 folds the content
// subtraction into A (f32 WMMA has no A-negate); the 1/9 quad scale is folded
// in too so the scatter needs no multiply.
// ---------------------------------------------------------------------------
__device__ __forceinline__ void setup_patch(const float* __restrict__ img,
                                            int b, int y, int x, float sgn,
                                            float* __restrict__ ot,
                                            float* __restrict__ od) {
    float px[9][3];
#pragma unroll
    for (int c = 0; c < 3; ++c) {
        const float* base = img + (((long)b * 3 + c) * HH + y) * WW + x;
#pragma unroll
        for (int k = 0; k < 9; ++k)
            px[k][c] = base[(k / 3) * WW + (k % 3)];
    }
    float s0 = 0.f, s1 = 0.f, s2 = 0.f;
    float c00 = 0.f, c01 = 0.f, c02 = 0.f, c11 = 0.f, c12 = 0.f, c22 = 0.f;
#pragma unroll
    for (int k = 0; k < 9; ++k) {
        float a = px[k][0], bb = px[k][1], c = px[k][2];
        s0 += a; s1 += bb; s2 += c;
        c00 += a * a; c01 += a * bb; c02 += a * c;
        c11 += bb * bb; c12 += bb * c; c22 += c * c;
    }
    const float inv9 = 1.0f / 9.0f;
    float m0 = s0 * inv9, m1 = s1 * inv9, m2 = s2 * inv9;
    c00 = c00 * inv9 - m0 * m0 + EPSF * inv9;
    c01 = c01 * inv9 - m0 * m1;
    c02 = c02 * inv9 - m0 * m2;
    c11 = c11 * inv9 - m1 * m1 + EPSF * inv9;
    c12 = c12 * inv9 - m1 * m2;
    c22 = c22 * inv9 - m2 * m2 + EPSF * inv9;
    // symmetric 3x3 inverse via adjugate; fold sgn and the final 1/9 scale in
    float a00 = c11 * c22 - c12 * c12;
    float a01 = c02 * c12 - c01 * c22;
    float a02 = c01 * c12 - c02 * c11;
    float det = c00 * a00 + c01 * a01 + c02 * a02;
    float id  = sgn / (9.0f * det);
    float i00 = a00 * id, i01 = a01 * id, i02 = a02 * id;
    float i11 = (c00 * c22 - c02 * c02) * id;
    float i12 = (c01 * c02 - c00 * c12) * id;
    float i22 = (c00 * c11 - c01 * c01) * id;
#pragma unroll
    for (int k = 0; k < 9; ++k) {
        float d0 = px[k][0] - m0, d1 = px[k][1] - m1, d2 = px[k][2] - m2;
        od[k * 3 + 0] = d0;
        od[k * 3 + 1] = d1;
        od[k * 3 + 2] = d2;
        ot[k * 3 + 0] = d0 * i00 + d1 * i01 + d2 * i02;
        ot[k * 3 + 1] = d0 * i01 + d1 * i11 + d2 * i12;
        ot[k * 3 + 2] = d0 * i02 + d1 * i12 + d2 * i22;
    }
}

// ---------------------------------------------------------------------------
// Pass 1: phase 1 stages 128 patches (thread-per-patch) into LDS; phase 2:
// each wave runs iters x { branchless LDS operand fetch, 2 chained
// V_WMMA_F32_16X16X4_F32, atomic scatter with uniform-base + constant
// per-lane offsets }.  Patch coords advance incrementally on SALU.
// LDS slot layout (floats): [0..26]=t_s  [27..53]=diff_s
//                           [54..80]=-t_c/9 [81..107]=diff_c
// ---------------------------------------------------------------------------
__global__ __launch_bounds__(TPB) void patch_wmma_kernel(
        const float* __restrict__ content,
        const float* __restrict__ stylized,
        float* __restrict__ wb /* [BATCH][NPIX][NBANDS] */) {
    __shared__ float lds[TPB * SLOT_STRIDE + 32];   // +tail pad for clamped reads
    const int tid  = threadIdx.x;
    const int base = (int)blockIdx.x * TPB;

    // ---- phase 1: thread-per-patch setup ----
    int pid = base + tid;
    if (pid < TOTALP) {
        int b = pid / NPATCH;
        int p = pid - b * NPATCH;
        int y = p / WP, x = p - y * WP;
        float* slot = &lds[tid * SLOT_STRIDE];
        setup_patch(stylized, b, y, x,  1.0f, slot,      slot + 27);
        setup_patch(content,  b, y, x, -1.0f, slot + 54, slot + 81);
    }
    __syncthreads();

    // ---- phase 2 prologue: loop-invariant per-lane state ----
    const int  lane = tid & 31;
    const int  wv   = tid >> 5;
    const int  mn   = lane & 15;
    const bool hi   = lane >= 16;
    const bool ok   = (mn < 9);
    const bool lo9  = ok && !hi;
    const int  mnc  = ok ? mn : 8;                 // clamp: pad rows read row 8 (finite junk)
    const int  o0   = mnc * 3 + (hi ? 2 : 0);      // .x element offset (K2 for hi lanes)
    const int  o1   = mnc * 3 + 1;                 // .y element offset (K1; K3 pad is zeroed)
    const int  ny = mn / 3, nx = mn % 3;
    int doff[8];                                   // (my*W+mx)*25 + band(M,N): loop-invariant
#pragma unroll
    for (int r = 0; r < 8; ++r) {
        int M = (r == 0) ? (hi ? 8 : 0) : r;       // C/D layout: lanes0-15 M=r, lanes16-31 M=r+8
        int my = M / 3, mx = M % 3;
        doff[r] = (my * WW + mx) * NBANDS + (ny - my + 2) * 5 + (nx - mx + 2);
    }

    // wave-uniform scalar state: trip count + incrementally advanced coords
    int sp0  = __builtin_amdgcn_readfirstlane(base + (wv << 5));
    int rem  = TOTALP - sp0;
    int iters = rem < 32 ? (rem < 0 ? 0 : rem) : 32;
    int b = sp0 / NPATCH;
    int pp = sp0 - b * NPATCH;
    int y = pp / WP, x = pp - y * WP;

    for (int j = 0; j < iters; ++j) {
        const float* slot = &lds[((wv << 5) + j) * SLOT_STRIDE];
        // branchless operand fetch (8 x ds_load_b32, pairable)
        float asx = slot[o0],      bsx = slot[27 + o0];
        float acx = slot[54 + o0], bcx = slot[81 + o0];
        float asy = slot[o1],      bsy = slot[27 + o1];
        float acy = slot[54 + o1], bcy = slot[81 + o1];
        v2f As = {asx, hi ? 0.f : asy};
        v2f Bs = {bsx, hi ? 0.f : bsy};
        v2f Ac = {acx, hi ? 0.f : acy};
        v2f Bc = {bcx, hi ? 0.f : bcy};

        v8f D = {};
        D = __builtin_amdgcn_wmma_f32_16x16x4_f32(false, As, false, Bs, (short)0, D, false, false);
        D = __builtin_amdgcn_wmma_f32_16x16x4_f32(false, Ac, false, Bc, (short)0, D, false, false);

        // scatter: uniform dword base + constant per-lane offsets
        int sbase = (b * NPIX + y * WW + x) * NBANDS;     // wave-uniform
        float* pb = wb + sbase;
        if (ok)  atomicAdd(pb + doff[0], D[0]);
        if (lo9) {
            atomicAdd(pb + doff[1], D[1]);
            atomicAdd(pb + doff[2], D[2]);
            atomicAdd(pb + doff[3], D[3]);
            atomicAdd(pb + doff[4], D[4]);
            atomicAdd(pb + doff[5], D[5]);
            atomicAdd(pb + doff[6], D[6]);
            atomicAdd(pb + doff[7], D[7]);
        }

        // advance patch coordinates incrementally (SALU, no division)
        if (++x == WP) { x = 0; if (++y == HP) { y = 0; ++b; } }
    }
}

// ---------------------------------------------------------------------------
// Pass 2: per pixel, fold band row into the squared-loss sum.
// L = -Wb with L[CENTER] += rowsum  =>  contrib = sum(w^2) - wC^2 + (d - wC)^2
// ---------------------------------------------------------------------------
__global__ __launch_bounds__(256) void reduce_kernel(const float* __restrict__ wb,
                                                     float* __restrict__ out) {
    const float inv_denom =
        (float)(1.0 / ((double)BATCH * (double)NPIX * (double)NPIX));
    long i = (long)blockIdx.x * blockDim.x + threadIdx.x;   // over BATCH*NPIX rows
    float contrib = 0.f;
    if (i < (long)BATCH * NPIX) {
        const float* w = wb + i * NBANDS;
        float s = 0.f, ss = 0.f;
#pragma unroll
        for (int k = 0; k < NBANDS; ++k) { float v = w[k]; s += v; ss += v * v; }
        float wc = w[CENTER];
        contrib = ss - wc * wc + (s - wc) * (s - wc);
    }
#pragma unroll
    for (int off = 16; off > 0; off >>= 1)
        contrib += __shfl_xor(contrib, off, 32);
    __shared__ float red[8];
    int lane = threadIdx.x & 31, wave = threadIdx.x >> 5;
    if (lane == 0) red[wave] = contrib;
    __syncthreads();
    if (threadIdx.x == 0) {
        float bs = 0.f;
#pragma unroll
        for (int wv = 0; wv < 8; ++wv) bs += red[wv];
        atomicAdd(out, bs * inv_denom);
    }
}

// ---------------------------------------------------------------------------
extern "C" void kernel_launch(void* const* d_in, const int* in_sizes, int n_in,
                              void* d_out, int out_size, void* d_ws, size_t ws_size,
                              hipStream_t stream) {
    (void)in_sizes; (void)n_in; (void)out_size; (void)ws_size;
    const float* content  = (const float*)d_in[0];
    const float* stylized = (const float*)d_in[1];
    float* out = (float*)d_out;
    float* wb  = (float*)d_ws;                       // BATCH*NPIX*NBANDS f32 = 26.2 MB

    long nzero = (long)BATCH * NPIX * NBANDS;
    zero_kernel<<<(int)((nzero + 255) / 256), 256, 0, stream>>>(wb, nzero, out);

    int blocks = (TOTALP + TPB - 1) / TPB;           // 2017
    patch_wmma_kernel<<<blocks, TPB, 0, stream>>>(content, stylized, wb);

    long rows = (long)BATCH * NPIX;                  // 262144
    reduce_kernel<<<(int)((rows + 255) / 256), 256, 0, stream>>>(wb, out);
}